// GridPooling_44753559224569
// MI455X (gfx1250) — compile-verified
//
#include <hip/hip_runtime.h>
#include <hip/hip_bf16.h>

#define GDIM 32
#define GC   (GDIM * GDIM * GDIM) /* 32768 cells */
#define FDIM 128                  /* features per point */
#define NB   8                    /* rows per async batch */
#define RWAVES 4                  /* waves per reduce block (128 threads) */

typedef float v4f __attribute__((ext_vector_type(4)));

__device__ __forceinline__ int cell_of(const float* __restrict__ pts, int i) {
    float x = pts[3 * i + 0];
    float y = pts[3 * i + 1];
    float z = pts[3 * i + 2];
    int ix = (int)floorf(x * (float)GDIM);
    int iy = (int)floorf(y * (float)GDIM);
    int iz = (int)floorf(z * (float)GDIM);
    ix = ix < 0 ? 0 : (ix > GDIM - 1 ? GDIM - 1 : ix);
    iy = iy < 0 ? 0 : (iy > GDIM - 1 ? GDIM - 1 : iy);
    iz = iz < 0 ? 0 : (iz > GDIM - 1 ? GDIM - 1 : iz);
    return ix * (GDIM * GDIM) + iy * GDIM + iz;
}

__global__ void gp_zero_i32(int* __restrict__ p, int n) {
    int i = blockIdx.x * blockDim.x + threadIdx.x;
    if (i < n) p[i] = 0;
}

__global__ void gp_count(const float* __restrict__ pts, int n, int* __restrict__ cnt) {
    int i = blockIdx.x * blockDim.x + threadIdx.x;
    if (i < n) atomicAdd(&cnt[cell_of(pts, i)], 1);
}

// One block of 1024 threads scans all 32768 counters: 32 serial elems per
// thread + LDS Hillis-Steele across the 1024 partial sums.
__global__ void gp_scan(const int* __restrict__ cnt, int* __restrict__ start,
                        int* __restrict__ cursor) {
    __shared__ int part[1024];
    const int t = threadIdx.x;
    int loc[32];
    int s = 0;
#pragma unroll
    for (int j = 0; j < 32; ++j) {
        loc[j] = cnt[t * 32 + j];
        s += loc[j];
    }
    part[t] = s;
    __syncthreads();
    for (int off = 1; off < 1024; off <<= 1) {
        int v = (t >= off) ? part[t - off] : 0;
        __syncthreads();
        part[t] += v;
        __syncthreads();
    }
    int excl = (t == 0) ? 0 : part[t - 1];
#pragma unroll
    for (int j = 0; j < 32; ++j) {
        start[t * 32 + j]  = excl;
        cursor[t * 32 + j] = excl;
        excl += loc[j];
    }
}

__global__ void gp_scatter(const float* __restrict__ pts, int n,
                           int* __restrict__ cursor, int* __restrict__ pidx) {
    int i = blockIdx.x * blockDim.x + threadIdx.x;
    if (i < n) {
        int c   = cell_of(pts, i);
        int pos = atomicAdd(&cursor[c], 1);
        pidx[pos] = i;
    }
}

// One wave32 per cell. Feature rows are streamed global->LDS with CDNA5
// async copies (GLOBAL_LOAD_ASYNC_TO_LDS_B128, ASYNCcnt), double-buffered in
// fixed batches of NB rows, then max-reduced out of LDS with unrolled
// ds_load_b128. Tail batches are padded by repeating the last point (max is
// idempotent). Single coalesced 512B store per cell; accumulator init 0
// implements the max(.,0) clamp and empty-cell zeroing.
__global__ void __launch_bounds__(RWAVES * 32)
gp_reduce(const float* __restrict__ feat, const int* __restrict__ pidx,
          const int* __restrict__ start, const int* __restrict__ cnt,
          float* __restrict__ out) {
    // 2 buffers x NB rows x 32 lanes of v4f per wave
    __shared__ v4f stage[RWAVES * 2 * NB * 32];

    const int gt   = blockIdx.x * blockDim.x + threadIdx.x;
    const int cell = gt >> 5;
    const int lane = gt & 31;
    const int wave = threadIdx.x >> 5;
    if (cell >= GC) return;

    v4f* my = &stage[wave * (2 * NB * 32)];

    const int s0 = start[cell];
    const int c  = cnt[cell];

    v4f m = (v4f){0.f, 0.f, 0.f, 0.f};

    if (c > 0) {
        const int nbatch = (c + NB - 1) / NB;

        auto issue_batch = [&](int b) {
            int ii = b * NB + lane;            // only lanes 0..NB-1 are used
            if (ii >= c) ii = c - 1;           // pad tail with last point
            int pv = pidx[s0 + ii];
            v4f* buf = my + (b & 1) * (NB * 32);
#pragma unroll
            for (int r = 0; r < NB; ++r) {
                int p = __shfl(pv, r, 32);
                unsigned long long ga =
                    (unsigned long long)(const void*)(feat + (size_t)p * FDIM) +
                    (unsigned long long)lane * 16u;
                unsigned la =
                    (unsigned)(unsigned long long)(const void*)&buf[r * 32 + lane];
                asm volatile("global_load_async_to_lds_b128 %0, %1, off"
                             :: "v"(la), "v"(ga) : "memory");
            }
        };

        issue_batch(0);
        for (int b = 0; b < nbatch; ++b) {
            if (b + 1 < nbatch) {
                issue_batch(b + 1);
                // in-order completion: <=NB outstanding => batch b landed
                asm volatile("s_wait_asynccnt 0x8" ::: "memory");
            } else {
                asm volatile("s_wait_asynccnt 0x0" ::: "memory");
            }
            const v4f* buf = my + (b & 1) * (NB * 32);
#pragma unroll
            for (int r = 0; r < NB; ++r) {
                v4f v = buf[r * 32 + lane];    // ds_load_b128, pipelined
                m.x = fmaxf(m.x, v.x);
                m.y = fmaxf(m.y, v.y);
                m.z = fmaxf(m.z, v.z);
                m.w = fmaxf(m.w, v.w);
            }
        }
    }
    ((v4f*)(out + (size_t)cell * FDIM))[lane] = m;  // also zeroes empty cells
}

// ---------------- fallback path (insufficient workspace) ----------------

__global__ void gp_zero_f32(float* __restrict__ p, int n) {
    int i = blockIdx.x * blockDim.x + threadIdx.x;
    if (i < n) p[i] = 0.f;
}

__global__ void gp_atomic(const float* __restrict__ feat, const float* __restrict__ pts,
                          int n, float* __restrict__ out) {
    int gt   = blockIdx.x * blockDim.x + threadIdx.x;
    int pt   = gt >> 5;
    int lane = gt & 31;
    if (pt >= n) return;
    int c = cell_of(pts, pt);
    v4f v = ((const v4f*)(feat + (size_t)pt * FDIM))[lane];
    int* o = (int*)(out + (size_t)c * FDIM + lane * 4);
    // out is 0-initialized; non-negative float bits order as signed ints,
    // and values <= 0 can never beat the init, so skip them.
    if (v.x > 0.f) atomicMax(o + 0, __float_as_int(v.x));
    if (v.y > 0.f) atomicMax(o + 1, __float_as_int(v.y));
    if (v.z > 0.f) atomicMax(o + 2, __float_as_int(v.z));
    if (v.w > 0.f) atomicMax(o + 3, __float_as_int(v.w));
}

extern "C" void kernel_launch(void* const* d_in, const int* in_sizes, int n_in,
                              void* d_out, int out_size, void* d_ws, size_t ws_size,
                              hipStream_t stream) {
    const float* feat = (const float*)d_in[0];
    const float* pts  = (const float*)d_in[1];
    float*       out  = (float*)d_out;
    const int n = in_sizes[1] / 3;

    const size_t need = (size_t)(3 * GC + n) * sizeof(int);

    if (d_ws != nullptr && ws_size >= need) {
        int* cnt    = (int*)d_ws;
        int* start  = cnt + GC;
        int* cursor = start + GC;
        int* pidx   = cursor + GC;

        gp_zero_i32<<<(GC + 255) / 256, 256, 0, stream>>>(cnt, GC);
        gp_count<<<(n + 255) / 256, 256, 0, stream>>>(pts, n, cnt);
        gp_scan<<<1, 1024, 0, stream>>>(cnt, start, cursor);
        gp_scatter<<<(n + 255) / 256, 256, 0, stream>>>(pts, n, cursor, pidx);
        gp_reduce<<<(GC * 32) / (RWAVES * 32), RWAVES * 32, 0, stream>>>(
            feat, pidx, start, cnt, out);
    } else {
        const int tot = GC * FDIM;
        gp_zero_f32<<<(tot + 255) / 256, 256, 0, stream>>>(out, tot);
        const long long threads = (long long)n * 32;
        gp_atomic<<<(unsigned)((threads + 255) / 256), 256, 0, stream>>>(feat, pts, n, out);
    }
}